// HistoryCentricModel_47863115547160
// MI455X (gfx1250) — compile-verified
//
#include <hip/hip_runtime.h>
#include <math.h>
#include <stdint.h>

// ---------------- problem constants ----------------
#define B_    512
#define S_    50
#define L_    40000
#define DM    80
#define DLOC  56
#define DUSR  12
#define H_    4
#define DH_   20
#define DFF   160
#define DK    160            // pred head inner dim
#define NT_N  2500           // 40000 / 16 column tiles
#define NT_M  32             // 512 / 16 row tiles
#define WPB   4              // waves (tiles) per gemm block
#define BLKN  (NT_N / WPB)   // 625
#define APAD  164            // LDS row stride for A tile (bank-conflict-free)

typedef float v2f __attribute__((ext_vector_type(2)));
typedef float v8f __attribute__((ext_vector_type(8)));

__device__ __forceinline__ float gelu_exact(float x) {
    return 0.5f * x * (1.0f + erff(x * 0.70710678118654752f));
}

// ---------------------------------------------------------------------------
// K1: token prep — embeddings + time features + tproj + input layernorm + pe
// one block per (b,s) token
// ---------------------------------------------------------------------------
__global__ void k_prep(const int* __restrict__ loc, const int* __restrict__ usr,
                       const int* __restrict__ wd, const float* __restrict__ smin,
                       const float* __restrict__ dur, const int* __restrict__ dif,
                       const float* __restrict__ loc_emb, const float* __restrict__ user_emb,
                       const float* __restrict__ tw, const float* __restrict__ tb,
                       const float* __restrict__ ng, const float* __restrict__ nb,
                       const float* __restrict__ pe, float* __restrict__ X)
{
    const int tok = blockIdx.x;
    const int s   = tok % S_;
    const int d   = threadIdx.x;
    __shared__ float xs[DM];
    __shared__ float te[12];
    __shared__ float st[2];

    const float TWO_PI = 6.283185307179586f;
    float tr = (smin[tok] / 60.0f) / 24.0f * TWO_PI;
    float wr = (float)wd[tok] / 7.0f * TWO_PI;
    float tf[6] = { __sinf(tr), __cosf(tr), log1pf(dur[tok]) / 8.0f,
                    __sinf(wr), __cosf(wr), (float)dif[tok] / 7.0f };
    if (d < 12) {
        float a = tb[d];
        #pragma unroll
        for (int i = 0; i < 6; ++i) a += tf[i] * tw[i * 12 + d];
        te[d] = a;
    }
    __syncthreads();
    if (d < DM) {
        float v;
        if (d < DLOC)            v = loc_emb[loc[tok] * DLOC + d];
        else if (d < DLOC + DUSR) v = user_emb[usr[tok] * DUSR + (d - DLOC)];
        else                     v = te[d - DLOC - DUSR];
        xs[d] = v;
    }
    __syncthreads();
    if (d == 0) {
        float m = 0.f;
        for (int i = 0; i < DM; ++i) m += xs[i];
        m /= DM;
        float v = 0.f;
        for (int i = 0; i < DM; ++i) { float t = xs[i] - m; v += t * t; }
        v /= DM;
        st[0] = m; st[1] = rsqrtf(v + 1e-5f);
    }
    __syncthreads();
    if (d < DM)
        X[tok * DM + d] = (xs[d] - st[0]) * st[1] * ng[d] + nb[d] + pe[s * DM + d];
}

// ---------------------------------------------------------------------------
// K2: qkv projection. K,V for all tokens; Q only at the last valid position.
// one block per token, 256 threads
// ---------------------------------------------------------------------------
__global__ void k_qkv(const float* __restrict__ X, const int* __restrict__ mask,
                      const float* __restrict__ w, const float* __restrict__ bq,
                      float* __restrict__ Qlast, float* __restrict__ Kb,
                      float* __restrict__ Vb)
{
    const int tok = blockIdx.x;
    const int b = tok / S_, s = tok % S_;
    __shared__ float xs[DM];
    __shared__ int lastIdx;
    const int j = threadIdx.x;
    if (j < DM) xs[j] = X[tok * DM + j];
    if (j == 0) {
        int c = 0;
        for (int t = 0; t < S_; ++t) c += (mask[b * S_ + t] != 0);
        lastIdx = c - 1;
    }
    __syncthreads();
    if (j < 3 * DM) {
        float a = bq[j];
        #pragma unroll 8
        for (int d = 0; d < DM; ++d) a += w[j * DM + d] * xs[d];
        if (j < DM)           { if (s == lastIdx) Qlast[b * DM + j] = a; }
        else if (j < 2 * DM)  Kb[tok * DM + (j - DM)] = a;
        else                  Vb[tok * DM + (j - 2 * DM)] = a;
    }
}

// ---------------------------------------------------------------------------
// K3: single-query attention per (b, head)
// ---------------------------------------------------------------------------
__global__ void k_attn(const float* __restrict__ Qlast, const float* __restrict__ Kb,
                       const float* __restrict__ Vb, const int* __restrict__ mask,
                       float* __restrict__ AO)
{
    const int b = blockIdx.x / H_, h = blockIdx.x % H_;
    const int t = threadIdx.x;
    __shared__ float q[DH_];
    __shared__ float p[S_];
    __shared__ float red[2];
    if (t < DH_) q[t] = Qlast[b * DM + h * DH_ + t];
    __syncthreads();
    if (t < S_) {
        float sc = -1e30f;
        if (mask[b * S_ + t] != 0) {
            sc = 0.f;
            #pragma unroll
            for (int i = 0; i < DH_; ++i) sc += q[i] * Kb[(b * S_ + t) * DM + h * DH_ + i];
            sc *= 0.22360679774997896f;   // 1/sqrt(20)
        }
        p[t] = sc;
    }
    __syncthreads();
    if (t == 0) {
        float m = -1e30f;
        for (int i = 0; i < S_; ++i) m = fmaxf(m, p[i]);
        red[0] = m;
    }
    __syncthreads();
    if (t < S_) p[t] = (p[t] > -1e29f) ? __expf(p[t] - red[0]) : 0.f;
    __syncthreads();
    if (t == 0) {
        float s = 0.f;
        for (int i = 0; i < S_; ++i) s += p[i];
        red[1] = 1.0f / s;
    }
    __syncthreads();
    if (t < DH_) {
        float a = 0.f;
        for (int i = 0; i < S_; ++i) a += p[i] * Vb[(b * S_ + i) * DM + h * DH_ + t];
        AO[b * DM + h * DH_ + t] = a * red[1];
    }
}

// ---------------------------------------------------------------------------
// K4: per-batch head: attn-out proj + res + ln1 + FFN + res + ln2 + pred1 gelu
// one block per b, 160 threads
// ---------------------------------------------------------------------------
__global__ void k_head(const float* __restrict__ X, const float* __restrict__ AO,
                       const int* __restrict__ mask,
                       const float* __restrict__ ow, const float* __restrict__ ob,
                       const float* __restrict__ g1, const float* __restrict__ b1,
                       const float* __restrict__ fw1, const float* __restrict__ fb1,
                       const float* __restrict__ fw2, const float* __restrict__ fb2,
                       const float* __restrict__ g2, const float* __restrict__ b2,
                       const float* __restrict__ pw1, const float* __restrict__ pb1,
                       float* __restrict__ H1)
{
    const int b = blockIdx.x;
    const int t = threadIdx.x;
    __shared__ float xl[DM], ao[DM], x1[DM], f1[DFF], x2[DM];
    __shared__ float st[2];
    __shared__ int li;
    if (t == 0) {
        int c = 0;
        for (int i = 0; i < S_; ++i) c += (mask[b * S_ + i] != 0);
        li = c - 1;
    }
    __syncthreads();
    if (t < DM) { xl[t] = X[(b * S_ + li) * DM + t]; ao[t] = AO[b * DM + t]; }
    __syncthreads();
    if (t < DM) {
        float a = ob[t];
        for (int d = 0; d < DM; ++d) a += ow[t * DM + d] * ao[d];
        x1[t] = xl[t] + a;
    }
    __syncthreads();
    if (t == 0) {
        float m = 0.f; for (int i = 0; i < DM; ++i) m += x1[i]; m /= DM;
        float v = 0.f; for (int i = 0; i < DM; ++i) { float d = x1[i] - m; v += d * d; }
        st[0] = m; st[1] = rsqrtf(v / DM + 1e-5f);
    }
    __syncthreads();
    if (t < DM) x1[t] = (x1[t] - st[0]) * st[1] * g1[t] + b1[t];
    __syncthreads();
    if (t < DFF) {
        float a = fb1[t];
        for (int d = 0; d < DM; ++d) a += x1[d] * fw1[d * DFF + t];
        f1[t] = gelu_exact(a);
    }
    __syncthreads();
    if (t < DM) {
        float a = fb2[t];
        for (int j = 0; j < DFF; ++j) a += f1[j] * fw2[j * DM + t];
        x2[t] = x1[t] + a;
    }
    __syncthreads();
    if (t == 0) {
        float m = 0.f; for (int i = 0; i < DM; ++i) m += x2[i]; m /= DM;
        float v = 0.f; for (int i = 0; i < DM; ++i) { float d = x2[i] - m; v += d * d; }
        st[0] = m; st[1] = rsqrtf(v / DM + 1e-5f);
    }
    __syncthreads();
    if (t < DM) x2[t] = (x2[t] - st[0]) * st[1] * g2[t] + b2[t];
    __syncthreads();
    if (t < DFF) {
        float a = pb1[t];
        for (int d = 0; d < DM; ++d) a += x2[d] * pw1[d * DFF + t];
        H1[b * DFF + t] = gelu_exact(a);
    }
}

// ---------------------------------------------------------------------------
// K5: big GEMM (512x40000x160) via V_WMMA_F32_16X16X4_F32.
// The 10KB A stripe (16 rows x 160 K, globally contiguous) is staged into LDS
// once per block with GLOBAL_LOAD_ASYNC_TO_LDS_B128 (+ s_wait_asynccnt), then
// shared by all 4 waves x 40 K-steps. LDS row stride padded to 164 floats so
// the 16 fragment lanes hit 16 distinct banks on ds_load_b64.
// mode 0: per-(row,tile) softmax partials (max, sumexp) -> ws
// mode 1: recompute, write model_weight * softmax * L into d_out
// block = 128 threads = 4 waves, each wave owns one 16x16 tile
// ---------------------------------------------------------------------------
__global__ void k_gemm(const float* __restrict__ A, const float* __restrict__ W2,
                       const float* __restrict__ bias, const float* __restrict__ rowMax,
                       const float* __restrict__ rowSum, const float* __restrict__ mwp,
                       float* __restrict__ partMax, float* __restrict__ partSum,
                       float* __restrict__ out, int mode)
{
    __shared__ float As[16 * APAD];
    const int tid    = threadIdx.x;
    const int wave   = tid >> 5;
    const int lane   = tid & 31;
    const int lane15 = lane & 15;
    const int hi     = lane >> 4;               // 0: lanes 0-15, 1: lanes 16-31
    const int mBase  = blockIdx.y * 16;
    const int nTile  = blockIdx.x * WPB + wave;
    const int nBase  = nTile * 16;

    // ---- async stage-in of the A stripe: 640 x 16B chunks over 128 threads
    {
        const float* Abase = A + (size_t)mBase * DK;
        #pragma unroll
        for (int i = 0; i < 5; ++i) {
            const int chunk = tid + i * 128;         // 0..639
            const int row = chunk / 40;              // 40 x 16B per 160-f32 row
            const int col = chunk - row * 40;        // 16B units within row
            unsigned long long ga =
                (unsigned long long)(Abase + row * DK + col * 4);
            unsigned ldsOff =
                (unsigned)(uintptr_t)(As + row * APAD + col * 4);
            asm volatile("global_load_async_to_lds_b128 %0, %1, off"
                         :: "v"(ldsOff), "v"(ga) : "memory");
        }
        asm volatile("s_wait_asynccnt 0x0" ::: "memory");
    }
    __syncthreads();

    v8f acc = {};
    #pragma unroll 8
    for (int k0 = 0; k0 < DK; k0 += 4) {
        const int ka = k0 + 2 * hi;
        // A fragment: 16x4 f32 (lane half selects K pair) — from LDS
        v2f a;
        a.x = As[lane15 * APAD + ka];
        a.y = As[lane15 * APAD + ka + 1];
        // B fragment: 4x16 f32, mirrored K layout, N striped across lanes
        v2f bf;
        bf.x = W2[(size_t)(ka)     * L_ + nBase + lane15];
        bf.y = W2[(size_t)(ka + 1) * L_ + nBase + lane15];
        if (k0 + 4 < DK)
            __builtin_prefetch(&W2[(size_t)(k0 + 4 + 2 * hi) * L_ + nBase + lane15], 0, 1);
        acc = __builtin_amdgcn_wmma_f32_16x16x4_f32(
            /*neg_a=*/false, a, /*neg_b=*/false, bf,
            /*c_mod=*/(short)0, acc, /*reuse_a=*/false, /*reuse_b=*/false);
    }

    const int   col = nBase + lane15;
    const float bb  = bias[col];
    float vals[8];
    #pragma unroll
    for (int r = 0; r < 8; ++r) vals[r] = acc[r] + bb;
    // C/D layout: VGPR r holds row mBase + r + 8*hi, col = nBase + (lane&15)

    if (mode == 0) {
        float m8[8], s8[8];
        #pragma unroll
        for (int r = 0; r < 8; ++r) {
            float m = vals[r];
            #pragma unroll
            for (int off = 1; off < 16; off <<= 1) m = fmaxf(m, __shfl_xor(m, off, 32));
            float e = __expf(vals[r] - m);
            #pragma unroll
            for (int off = 1; off < 16; off <<= 1) e += __shfl_xor(e, off, 32);
            m8[r] = m; s8[r] = e;
        }
        if (lane15 == 0) {
            #pragma unroll
            for (int r = 0; r < 8; ++r) {
                const int row = mBase + r + 8 * hi;
                partMax[row * NT_N + nTile] = m8[r];
                partSum[row * NT_N + nTile] = s8[r];
            }
        }
    } else {
        const float mw = mwp[0] * (float)L_;
        #pragma unroll
        for (int r = 0; r < 8; ++r) {
            const int row = mBase + r + 8 * hi;
            const float p = __expf(vals[r] - rowMax[row]) / rowSum[row];
            out[(size_t)row * L_ + col] = mw * p;
        }
    }
}

// ---------------------------------------------------------------------------
// K6: merge softmax partials per row (log-sum-exp merge)
// ---------------------------------------------------------------------------
__global__ void k_reduce(const float* __restrict__ partMax, const float* __restrict__ partSum,
                         float* __restrict__ rowMax, float* __restrict__ rowSum)
{
    const int row = blockIdx.x;
    const int t = threadIdx.x;
    __shared__ float sm[256];
    float m = -1e30f;
    for (int i = t; i < NT_N; i += 256) m = fmaxf(m, partMax[row * NT_N + i]);
    sm[t] = m; __syncthreads();
    for (int off = 128; off > 0; off >>= 1) {
        if (t < off) sm[t] = fmaxf(sm[t], sm[t + off]);
        __syncthreads();
    }
    const float M = sm[0];
    __syncthreads();
    float s = 0.f;
    for (int i = t; i < NT_N; i += 256)
        s += partSum[row * NT_N + i] * __expf(partMax[row * NT_N + i] - M);
    sm[t] = s; __syncthreads();
    for (int off = 128; off > 0; off >>= 1) {
        if (t < off) sm[t] += sm[t + off];
        __syncthreads();
    }
    if (t == 0) { rowMax[row] = M; rowSum[row] = sm[0]; }
}

// ---------------------------------------------------------------------------
// K7: history scores — O(S^2) per row; first-occurrence guard => no atomics
// ---------------------------------------------------------------------------
__global__ void k_hist(const int* __restrict__ loc, const int* __restrict__ mask,
                       const float* __restrict__ decay_p, const float* __restrict__ fw_p,
                       const float* __restrict__ hs_p, float* __restrict__ out)
{
    const int b = blockIdx.x;
    const int t = threadIdx.x;
    __shared__ float rw[S_];
    __shared__ float fr[S_];
    __shared__ int   lc[S_], mk[S_];
    __shared__ float mf;
    const float decay = decay_p[0];
    if (t < S_) {
        lc[t] = loc[b * S_ + t];
        mk[t] = (mask[b * S_ + t] != 0);
        rw[t] = mk[t] ? powf(decay, (float)(S_ - 1 - t)) : 0.f;
    }
    __syncthreads();
    float freq = 0.f, rec = 0.f;
    int first = 0;
    if (t < S_ && mk[t]) {
        first = 1;
        for (int u = 0; u < S_; ++u) {
            if (mk[u] && lc[u] == lc[t]) {
                freq += 1.f;
                rec = fmaxf(rec, rw[u]);
                if (u < t) first = 0;
            }
        }
    }
    if (t < S_) fr[t] = mk[t] ? freq : 0.f;
    __syncthreads();
    if (t == 0) {
        float m = 0.f;
        for (int i = 0; i < S_; ++i) m = fmaxf(m, fr[i]);
        mf = fmaxf(m, 1.0f);
    }
    __syncthreads();
    if (t < S_ && first) {
        const float v = hs_p[0] * (rec + fw_p[0] * freq / mf);
        out[(size_t)b * L_ + lc[t]] += v;
    }
}

// ---------------------------------------------------------------------------
extern "C" void kernel_launch(void* const* d_in, const int* in_sizes, int n_in,
                              void* d_out, int out_size, void* d_ws, size_t ws_size,
                              hipStream_t stream) {
    const int*   loc      = (const int*)d_in[0];
    const int*   usr      = (const int*)d_in[1];
    const int*   wd       = (const int*)d_in[2];
    const float* smin     = (const float*)d_in[3];
    const float* dur      = (const float*)d_in[4];
    const int*   dif      = (const int*)d_in[5];
    const int*   mask     = (const int*)d_in[6];
    const float* loc_emb  = (const float*)d_in[7];
    const float* user_emb = (const float*)d_in[8];
    const float* tw       = (const float*)d_in[9];
    const float* tb       = (const float*)d_in[10];
    const float* ing      = (const float*)d_in[11];
    const float* inb      = (const float*)d_in[12];
    const float* aiw      = (const float*)d_in[13];
    const float* aib      = (const float*)d_in[14];
    const float* aow      = (const float*)d_in[15];
    const float* aob      = (const float*)d_in[16];
    const float* n1g      = (const float*)d_in[17];
    const float* n1b      = (const float*)d_in[18];
    const float* fw1      = (const float*)d_in[19];
    const float* fb1      = (const float*)d_in[20];
    const float* fw2      = (const float*)d_in[21];
    const float* fb2      = (const float*)d_in[22];
    const float* n2g      = (const float*)d_in[23];
    const float* n2b      = (const float*)d_in[24];
    const float* pw1      = (const float*)d_in[25];
    const float* pb1      = (const float*)d_in[26];
    const float* pw2      = (const float*)d_in[27];
    const float* pb2      = (const float*)d_in[28];
    const float* decay    = (const float*)d_in[29];
    const float* fwgt     = (const float*)d_in[30];
    const float* hsc      = (const float*)d_in[31];
    const float* mwt      = (const float*)d_in[32];
    const float* pe       = (const float*)d_in[33];

    float* ws = (float*)d_ws;
    float* X       = ws;                         // B*S*80
    float* Kb      = X    + (size_t)B_ * S_ * DM;
    float* Vb      = Kb   + (size_t)B_ * S_ * DM;
    float* Qlast   = Vb   + (size_t)B_ * S_ * DM;  // B*80
    float* AO      = Qlast + (size_t)B_ * DM;      // B*80
    float* H1      = AO   + (size_t)B_ * DM;       // B*160
    float* partMax = H1   + (size_t)B_ * DFF;      // 512*2500
    float* partSum = partMax + (size_t)B_ * NT_N;
    float* rowMax  = partSum + (size_t)B_ * NT_N;  // 512
    float* rowSum  = rowMax + B_;

    float* out = (float*)d_out;

    k_prep<<<B_ * S_, 128, 0, stream>>>(loc, usr, wd, smin, dur, dif,
                                        loc_emb, user_emb, tw, tb, ing, inb, pe, X);
    k_qkv<<<B_ * S_, 256, 0, stream>>>(X, mask, aiw, aib, Qlast, Kb, Vb);
    k_attn<<<B_ * H_, 64, 0, stream>>>(Qlast, Kb, Vb, mask, AO);
    k_head<<<B_, DFF, 0, stream>>>(X, AO, mask, aow, aob, n1g, n1b,
                                   fw1, fb1, fw2, fb2, n2g, n2b, pw1, pb1, H1);
    dim3 gg(BLKN, NT_M);
    k_gemm<<<gg, 128, 0, stream>>>(H1, pw2, pb2, rowMax, rowSum, mwt,
                                   partMax, partSum, out, /*mode=*/0);
    k_reduce<<<B_, 256, 0, stream>>>(partMax, partSum, rowMax, rowSum);
    k_gemm<<<gg, 128, 0, stream>>>(H1, pw2, pb2, rowMax, rowSum, mwt,
                                   partMax, partSum, out, /*mode=*/1);
    k_hist<<<B_, 64, 0, stream>>>(loc, mask, decay, fwgt, hsc, out);
}